// FastMultiEMAMixture_21285857919027
// MI455X (gfx1250) — compile-verified
//
#include <hip/hip_runtime.h>
#include <stdint.h>

static constexpr int   Bn   = 16;
static constexpr int   Tn   = 4096;
static constexpr int   Cn   = 512;
static constexpr int   Kn   = 3;
static constexpr float EPSC = 1e-4f;

static constexpr int NCHUNK   = 64;             // chunks along T
static constexpr int LC       = Tn / NCHUNK;    // 64 timesteps per chunk
static constexpr int ROWS     = 8;              // timesteps per async pipeline stage
static constexpr int NSTAGE   = LC / ROWS;      // 8 stages per chunk
static constexpr int NTHREADS = Cn / 4;         // 128 threads, 4 channels (float4) each

static constexpr uint32_t ROW_BYTES   = (uint32_t)NTHREADS * 16u;   // 2048 (== Cn*4)
static constexpr uint32_t STAGE_BYTES = (uint32_t)ROWS * ROW_BYTES; // 16384

static_assert(LC == 64, "a^LC computed with 6 squarings");
static_assert(ROW_BYTES == (uint32_t)Cn * 4u, "LDS row stride == global row stride");

// Native clang vector type (required by __builtin_nontemporal_store).
typedef float v4f __attribute__((ext_vector_type(4)));

// ---------------------------------------------------------------------------
// CDNA5 async copy: per-lane 16B global -> LDS, tracked with ASYNCcnt.
// GVS mode: 64-bit uniform SGPR base + 32-bit per-lane byte offset in VGPR.
// ---------------------------------------------------------------------------
__device__ __forceinline__ void async_ld_b128(uint32_t lds_byte, const void* gbase,
                                              uint32_t gbyte) {
  asm volatile("global_load_async_to_lds_b128 %0, %1, %2"
               :
               : "v"(lds_byte), "v"(gbyte), "s"(gbase)
               : "memory");
}

template <int N>
__device__ __forceinline__ void wait_asynccnt() {
  asm volatile("s_wait_asynccnt %0" : : "n"(N) : "memory");
}

__device__ __forceinline__ float4 ema4(float4 a, float4 y, float4 b, float4 x) {
  float4 r;
  r.x = fmaf(a.x, y.x, b.x * x.x);
  r.y = fmaf(a.y, y.y, b.y * x.y);
  r.z = fmaf(a.z, y.z, b.z * x.z);
  r.w = fmaf(a.w, y.w, b.w * x.w);
  return r;
}

// ---------------------------------------------------------------------------
// Kernel 0: per-channel parameters. a = clip(sigmoid(logit_alpha)), b = 1-a,
// mix = softmax_K(mix_logits). Stored transposed to [K][C] for coalescing.
// ---------------------------------------------------------------------------
__global__ void ema_params(const float* __restrict__ logit_alpha,
                           const float* __restrict__ mix_logits,
                           float* __restrict__ aP, float* __restrict__ bP,
                           float* __restrict__ mP) {
  int c = blockIdx.x * blockDim.x + threadIdx.x;
  if (c >= Cn) return;
  float m0 = mix_logits[c * Kn + 0];
  float m1 = mix_logits[c * Kn + 1];
  float m2 = mix_logits[c * Kn + 2];
  float mx = fmaxf(m0, fmaxf(m1, m2));
  float e0 = expf(m0 - mx), e1 = expf(m1 - mx), e2 = expf(m2 - mx);
  float inv = 1.0f / (e0 + e1 + e2);
  mP[0 * Cn + c] = e0 * inv;
  mP[1 * Cn + c] = e1 * inv;
  mP[2 * Cn + c] = e2 * inv;
#pragma unroll
  for (int k = 0; k < Kn; ++k) {
    float a = 1.0f / (1.0f + expf(-logit_alpha[k * Cn + c]));
    a = fminf(fmaxf(a, EPSC), 1.0f - EPSC);
    aP[k * Cn + c] = a;
    bP[k * Cn + c] = 1.0f - a;
  }
}

// ---------------------------------------------------------------------------
// Kernel 1: chunk-local EMA end-states (zero init; chunk 0 uses y[0]=x[0]).
// summ[b][chunk][k][c] = local end state u_j.
// ---------------------------------------------------------------------------
__global__ __launch_bounds__(NTHREADS) void ema_partial(
    const float* __restrict__ x, const float* __restrict__ aP,
    const float* __restrict__ bP, float* __restrict__ summ) {
  __shared__ __align__(16) float4 sm[2 * ROWS * NTHREADS];
  const int tid = threadIdx.x;
  const int chunk = blockIdx.x;
  const int b = blockIdx.y;

  const float4* a4 = (const float4*)aP;
  const float4* b4 = (const float4*)bP;
  float4 A0 = a4[tid], A1 = a4[NTHREADS + tid], A2 = a4[2 * NTHREADS + tid];
  float4 Bv0 = b4[tid], Bv1 = b4[NTHREADS + tid], Bv2 = b4[2 * NTHREADS + tid];

  const uint32_t g0 =
      (uint32_t)((((size_t)b * Tn + (size_t)chunk * LC) * Cn + 4 * (size_t)tid) * 4);
  const uint32_t lds0 = (uint32_t)(uintptr_t)(&sm[0]) + (uint32_t)tid * 16u;

#pragma unroll
  for (int r = 0; r < ROWS; ++r)  // prologue: stage 0
    async_ld_b128(lds0 + (uint32_t)r * ROW_BYTES, x, g0 + (uint32_t)r * ROW_BYTES);

  float4 s0 = make_float4(0.f, 0.f, 0.f, 0.f), s1 = s0, s2 = s0;
  const int tbase = chunk * LC;

  for (int s = 0; s < NSTAGE; ++s) {
    if (s + 1 < NSTAGE) {
      const uint32_t gs = g0 + (uint32_t)(s + 1) * STAGE_BYTES;
      const uint32_t ls = lds0 + (uint32_t)((s + 1) & 1) * STAGE_BYTES;
#pragma unroll
      for (int r = 0; r < ROWS; ++r)
        async_ld_b128(ls + (uint32_t)r * ROW_BYTES, x, gs + (uint32_t)r * ROW_BYTES);
      wait_asynccnt<ROWS>();  // stage s complete; stage s+1 (8 ops) in flight
    } else {
      wait_asynccnt<0>();
    }
    const float4* buf = &sm[(s & 1) * ROWS * NTHREADS];
#pragma unroll
    for (int r = 0; r < ROWS; ++r) {
      float4 xv = buf[r * NTHREADS + tid];
      int t = tbase + s * ROWS + r;
      if (t == 0) {
        s0 = xv; s1 = xv; s2 = xv;  // y[0] = x[0] for all k
      } else {
        s0 = ema4(A0, s0, Bv0, xv);
        s1 = ema4(A1, s1, Bv1, xv);
        s2 = ema4(A2, s2, Bv2, xv);
      }
    }
  }
  float4* o4 = (float4*)summ;
  size_t base = (size_t)(b * NCHUNK + chunk) * Kn * NTHREADS;
  o4[base + tid] = s0;
  o4[base + NTHREADS + tid] = s1;
  o4[base + 2 * NTHREADS + tid] = s2;
}

// ---------------------------------------------------------------------------
// Kernel 2: propagate exact end-states across chunks in place:
//   e_0 = u_0 ;  e_j = a^LC * e_{j-1} + u_j
// ---------------------------------------------------------------------------
__global__ __launch_bounds__(128) void ema_carry(float* __restrict__ summ,
                                                 const float* __restrict__ aP) {
  int idx = blockIdx.x * 128 + threadIdx.x;
  if (idx >= Bn * Kn * NTHREADS) return;
  int c4 = idx % NTHREADS;
  int k = (idx / NTHREADS) % Kn;
  int b = idx / (NTHREADS * Kn);

  float4 aL = ((const float4*)aP)[k * NTHREADS + c4];
#pragma unroll
  for (int i = 0; i < 6; ++i) {  // a^(2^6) == a^LC
    aL.x *= aL.x; aL.y *= aL.y; aL.z *= aL.z; aL.w *= aL.w;
  }
  float4* s4 = (float4*)summ;
  const size_t stride = (size_t)Kn * NTHREADS;  // per-chunk stride in float4
  size_t base = (size_t)b * NCHUNK * stride + (size_t)k * NTHREADS + c4;
  float4 e = s4[base];
#pragma unroll 4
  for (int j = 1; j < NCHUNK; ++j) {
    float4 u = s4[base + (size_t)j * stride];
    e.x = fmaf(aL.x, e.x, u.x);
    e.y = fmaf(aL.y, e.y, u.y);
    e.z = fmaf(aL.z, e.z, u.z);
    e.w = fmaf(aL.w, e.w, u.w);
    s4[base + (size_t)j * stride] = e;  // now holds exact end state e_j
  }
}

// ---------------------------------------------------------------------------
// Kernel 3: seed with carry e_{chunk-1}, run the reference recurrence over the
// chunk, mix over K, and stream out with non-temporal stores (preserve x in L2).
// ---------------------------------------------------------------------------
__global__ __launch_bounds__(NTHREADS) void ema_final(
    const float* __restrict__ x, const float* __restrict__ aP,
    const float* __restrict__ bP, const float* __restrict__ mP,
    const float* __restrict__ summ, float* __restrict__ out) {
  __shared__ __align__(16) float4 sm[2 * ROWS * NTHREADS];
  const int tid = threadIdx.x;
  const int chunk = blockIdx.x;
  const int b = blockIdx.y;

  const float4* a4 = (const float4*)aP;
  const float4* b4 = (const float4*)bP;
  const float4* m4 = (const float4*)mP;
  float4 A0 = a4[tid], A1 = a4[NTHREADS + tid], A2 = a4[2 * NTHREADS + tid];
  float4 Bv0 = b4[tid], Bv1 = b4[NTHREADS + tid], Bv2 = b4[2 * NTHREADS + tid];
  float4 M0 = m4[tid], M1 = m4[NTHREADS + tid], M2 = m4[2 * NTHREADS + tid];

  float4 y0, y1, y2;
  if (chunk == 0) {
    y0 = make_float4(0.f, 0.f, 0.f, 0.f); y1 = y0; y2 = y0;  // overwritten at t==0
  } else {
    const float4* c4p = (const float4*)summ;
    size_t base = (size_t)(b * NCHUNK + chunk - 1) * Kn * NTHREADS;
    y0 = c4p[base + tid];
    y1 = c4p[base + NTHREADS + tid];
    y2 = c4p[base + 2 * NTHREADS + tid];
  }

  const uint32_t g0 =
      (uint32_t)((((size_t)b * Tn + (size_t)chunk * LC) * Cn + 4 * (size_t)tid) * 4);
  const uint32_t lds0 = (uint32_t)(uintptr_t)(&sm[0]) + (uint32_t)tid * 16u;

#pragma unroll
  for (int r = 0; r < ROWS; ++r)
    async_ld_b128(lds0 + (uint32_t)r * ROW_BYTES, x, g0 + (uint32_t)r * ROW_BYTES);

  const int tbase = chunk * LC;
  v4f* outp = (v4f*)out;

  for (int s = 0; s < NSTAGE; ++s) {
    if (s + 1 < NSTAGE) {
      const uint32_t gs = g0 + (uint32_t)(s + 1) * STAGE_BYTES;
      const uint32_t ls = lds0 + (uint32_t)((s + 1) & 1) * STAGE_BYTES;
#pragma unroll
      for (int r = 0; r < ROWS; ++r)
        async_ld_b128(ls + (uint32_t)r * ROW_BYTES, x, gs + (uint32_t)r * ROW_BYTES);
      wait_asynccnt<ROWS>();
    } else {
      wait_asynccnt<0>();
    }
    const float4* buf = &sm[(s & 1) * ROWS * NTHREADS];
#pragma unroll
    for (int r = 0; r < ROWS; ++r) {
      float4 xv = buf[r * NTHREADS + tid];
      int t = tbase + s * ROWS + r;
      if (t == 0) {
        y0 = xv; y1 = xv; y2 = xv;
      } else {
        y0 = ema4(A0, y0, Bv0, xv);
        y1 = ema4(A1, y1, Bv1, xv);
        y2 = ema4(A2, y2, Bv2, xv);
      }
      v4f o;
      o.x = fmaf(M2.x, y2.x, fmaf(M1.x, y1.x, M0.x * y0.x));
      o.y = fmaf(M2.y, y2.y, fmaf(M1.y, y1.y, M0.y * y0.y));
      o.z = fmaf(M2.z, y2.z, fmaf(M1.z, y1.z, M0.z * y0.z));
      o.w = fmaf(M2.w, y2.w, fmaf(M1.w, y1.w, M0.w * y0.w));
      __builtin_nontemporal_store(o, outp + ((size_t)b * Tn + t) * (Cn / 4) + tid);
    }
  }
}

// ---------------------------------------------------------------------------
extern "C" void kernel_launch(void* const* d_in, const int* in_sizes, int n_in,
                              void* d_out, int out_size, void* d_ws, size_t ws_size,
                              hipStream_t stream) {
  (void)in_sizes; (void)n_in; (void)out_size; (void)ws_size;
  const float* x           = (const float*)d_in[0];  // [B,T,C]
  const float* logit_alpha = (const float*)d_in[1];  // [K,C]
  const float* mix_logits  = (const float*)d_in[2];  // [C,K]
  float* out = (float*)d_out;                        // [B,T,C]

  float* aP = (float*)d_ws;          // [K,C]
  float* bP = aP + Kn * Cn;          // [K,C]
  float* mP = bP + Kn * Cn;          // [K,C]
  float* summ = mP + Kn * Cn;        // [B,NCHUNK,K,C]  (~6.3 MB)

  ema_params<<<(Cn + 127) / 128, 128, 0, stream>>>(logit_alpha, mix_logits, aP, bP, mP);
  ema_partial<<<dim3(NCHUNK, Bn), NTHREADS, 0, stream>>>(x, aP, bP, summ);
  ema_carry<<<(Bn * Kn * NTHREADS + 127) / 128, 128, 0, stream>>>(summ, aP);
  ema_final<<<dim3(NCHUNK, Bn), NTHREADS, 0, stream>>>(x, aP, bP, mP, summ, out);
}